// iSKDE_28157805593443
// MI455X (gfx1250) — compile-verified
//
#include <hip/hip_runtime.h>

typedef __attribute__((ext_vector_type(2))) float v2f;
typedef __attribute__((ext_vector_type(8))) float v8f;

#define HEADS 8
#define MHDIM 8
#define FEAT  (HEADS * MHDIM)   // 64 floats per query row

// One wave32 handles 16 queries. For head h:
//   A (16x4 f32): rows = queries, cols = 4 features of head h      (2 VGPRs)
//   B (4x16 f32): rows = 4 features, cols = gathered node columns  (2 VGPRs)
//   D = A x B accumulated over 2 k-steps (MH=8); diagonal D[q][q] = per-query dot.
// Head loop is intentionally NOT unrolled: keeps one (cp,cn) pair live -> low
// VGPR count -> high occupancy to hide the random-gather latency.
__global__ __launch_bounds__(256) void iskde_wmma_kernel(
    const float* __restrict__ q_phi,     // [n_q, 64]
    const float* __restrict__ pk_pos,    // [64, n_nodes]  (h*8+mh major)
    const float* __restrict__ pk_neg,    // [64, n_nodes]
    const float* __restrict__ deg_pos,   // [n_nodes, 8]
    const float* __restrict__ deg_neg,   // [n_nodes, 8]
    const int*   __restrict__ k_idx,     // [n_q]
    float* __restrict__ out_pos,         // [n_q, 8]
    float* __restrict__ out_neg,         // [n_q, 8]
    int n_q, int n_nodes)
{
    const int lane = threadIdx.x & 31;
    const int wave = threadIdx.x >> 5;
    const int tile = blockIdx.x * (blockDim.x >> 5) + wave;   // 16 queries / tile
    const int qbase = tile * 16;
    if (qbase >= n_q) return;            // wave-uniform: EXEC stays all-1s below

    const int m  = lane & 15;            // query-in-tile (rows of A / cols of B / lanes of D)
    const int hi = lane >> 4;            // half-wave selects K pairs {0,1} vs {2,3}
    const int q  = qbase + m;
    const int qc = (q < n_q) ? q : (n_q - 1);          // clamp tail for loads

    // node for column m (lanes 16-31 mirror lanes 0-15, as B layout requires)
    const int node = k_idx[qc];

    const float* __restrict__ qrow = q_phi + (size_t)qc * FEAT;
    const bool writer = (lane < 16) && (q < n_q);

#pragma unroll 1
    for (int h = 0; h < HEADS; ++h) {
        v8f cp = {};
        v8f cn = {};
#pragma unroll
        for (int s = 0; s < 2; ++s) {
            const int f = h * MHDIM + s * 4;           // base feature of this k-step
            // A 16x4 layout: VGPR0 = {K0 | K2}, VGPR1 = {K1 | K3} across lane halves.
            // q_phi is streamed exactly once -> non-temporal, keep L2 for the gathers.
            v2f a;
            a.x = __builtin_nontemporal_load(qrow + f + 2 * hi);
            a.y = __builtin_nontemporal_load(qrow + f + 2 * hi + 1);

            // B 4x16 (transposed gather): lane col = m, K split across halves like A.
            const size_t off0 = (size_t)(f + 2 * hi)     * (size_t)n_nodes + (size_t)node;
            const size_t off1 = (size_t)(f + 2 * hi + 1) * (size_t)n_nodes + (size_t)node;
            v2f bp, bn;
            bp.x = pk_pos[off0];  bp.y = pk_pos[off1];  // RT: want these resident in L2
            bn.x = pk_neg[off0];  bn.y = pk_neg[off1];

            cp = __builtin_amdgcn_wmma_f32_16x16x4_f32(false, a, false, bp,
                                                       (short)0, cp, false, false);
            cn = __builtin_amdgcn_wmma_f32_16x16x4_f32(false, a, false, bn,
                                                       (short)0, cn, false, false);
        }

        // Diagonal extraction from D (16x16 f32 C/D layout):
        //   vgpr r, lane r      -> D[r][r]
        //   vgpr r, lane r+24   -> D[r+8][r+8]
        float rp = 0.f, rn = 0.f;
#pragma unroll
        for (int r = 0; r < 8; ++r) {
            const float p0 = __shfl(cp[r], r,      32);
            const float p8 = __shfl(cp[r], r + 24, 32);
            const float n0 = __shfl(cn[r], r,      32);
            const float n8 = __shfl(cn[r], r + 24, 32);
            if (m == r)     { rp = p0; rn = n0; }
            if (m == r + 8) { rp = p8; rn = n8; }
        }

        float dp = deg_pos[(size_t)node * HEADS + h];
        float dn = deg_neg[(size_t)node * HEADS + h];
        dp = dp > 1.0f ? dp : 1.0f;
        dn = dn > 1.0f ? dn : 1.0f;

        // Store this head's result immediately (no live accumulator arrays).
        // L2 write-merging assembles the 32B rows into full lines.
        if (writer) {
            out_pos[(size_t)q * HEADS + h] = rp / dp;
            out_neg[(size_t)q * HEADS + h] = rn / dn;
        }
    }
}

extern "C" void kernel_launch(void* const* d_in, const int* in_sizes, int n_in,
                              void* d_out, int out_size, void* d_ws, size_t ws_size,
                              hipStream_t stream)
{
    (void)n_in; (void)out_size; (void)d_ws; (void)ws_size;

    const float* q_phi   = (const float*)d_in[0];
    const float* pk_pos  = (const float*)d_in[1];
    const float* pk_neg  = (const float*)d_in[2];
    const float* deg_pos = (const float*)d_in[3];
    const float* deg_neg = (const float*)d_in[4];
    const int*   k_idx   = (const int*)  d_in[5];

    const int n_q     = in_sizes[5];               // k_indices is [N_Q]
    const int n_nodes = in_sizes[3] / HEADS;       // deg_pos is [N_NODES, H]

    float* out_pos = (float*)d_out;
    float* out_neg = out_pos + (size_t)n_q * HEADS;

    const int tiles          = (n_q + 15) / 16;    // 16 queries per wave
    const int waves_per_blk  = 8;                  // 256 threads = 8 wave32
    const int blocks         = (tiles + waves_per_blk - 1) / waves_per_blk;

    iskde_wmma_kernel<<<blocks, waves_per_blk * 32, 0, stream>>>(
        q_phi, pk_pos, pk_neg, deg_pos, deg_neg, k_idx,
        out_pos, out_neg, n_q, n_nodes);
}